// GraphSAGE_node_classfication_gaussion_77180562309273
// MI455X (gfx1250) — compile-verified
//
#include <hip/hip_runtime.h>

typedef __attribute__((ext_vector_type(2))) float v2f;
typedef __attribute__((ext_vector_type(8))) float v8f;

#define GN 40000
#define GE 640000
#define GDH 128

// ---------------- CSR build ----------------

__global__ void count_kernel(const int* __restrict__ dst, int* __restrict__ cnt, int E) {
    int e = blockIdx.x * blockDim.x + threadIdx.x;
    if (e < E) atomicAdd(&cnt[dst[e]], 1);
}

__global__ void scan_kernel(const int* __restrict__ cnt, int* __restrict__ row_ptr,
                            float* __restrict__ deg_inv, int n) {
    __shared__ int ssum[1024];
    int t = threadIdx.x;
    int chunk = (n + 1023) / 1024;
    int beg = t * chunk;
    int end = beg + chunk; if (end > n) end = n;
    int s = 0;
    for (int i = beg; i < end; ++i) s += cnt[i];
    ssum[t] = s;
    __syncthreads();
    for (int off = 1; off < 1024; off <<= 1) {
        int v = (t >= off) ? ssum[t - off] : 0;
        __syncthreads();
        ssum[t] += v;
        __syncthreads();
    }
    int run = (t == 0) ? 0 : ssum[t - 1];
    for (int i = beg; i < end; ++i) {
        row_ptr[i] = run;
        int c = cnt[i];
        run += c;
        deg_inv[i] = 1.0f / (float)(c > 0 ? c : 1);
    }
    if (t == 1023) row_ptr[n] = run;
}

__global__ void fill_kernel(const int* __restrict__ src, const int* __restrict__ dst,
                            const int* __restrict__ row_ptr, int* __restrict__ fill,
                            int* __restrict__ csr_src, int E) {
    int e = blockIdx.x * blockDim.x + threadIdx.x;
    if (e < E) {
        int d = dst[e];
        int p = atomicAdd(&fill[d], 1);
        csr_src[row_ptr[d] + p] = src[e];
    }
}

// ---------------- Weight transpose with zero padding ----------------
// wt[c*K + k] = (c < Dout) ? w[k*Dout + c] : 0   for c in [0, DoutPad)

__global__ void transpose_pad(const float* __restrict__ w, float* __restrict__ wt,
                              int K, int Dout, int DoutPad) {
    int idx = blockIdx.x * blockDim.x + threadIdx.x;
    int total = DoutPad * K;
    if (idx >= total) return;
    int c = idx / K;
    int k = idx - c * K;
    wt[idx] = (c < Dout) ? w[(size_t)k * Dout + c] : 0.0f;
}

// ---------------- Dual GEMM via V_WMMA_F32_16X16X4_F32 ----------------
// Z = A @ Wl ; R = A @ Wr + bias.  B operands come from transposed, padded
// weights so every load is an unguarded aligned b64 with constant stride.
// Each wave computes MT vertical 16x16 tiles sharing one B fetch per k-step.

template<int DOUT, int DPAD, int MT>
__global__ void sage_gemm_dual(const float* __restrict__ A,     // M x K
                               const float* __restrict__ WlT,   // DPAD x K (transposed)
                               const float* __restrict__ WrT,   // DPAD x K (transposed)
                               const float* __restrict__ bias,  // DOUT
                               float* __restrict__ Z, float* __restrict__ R,
                               int M) {
    constexpr int K  = GDH;
    constexpr int NT = DPAD / 16;
    int wave = blockIdx.x * blockDim.y + threadIdx.y;
    int mg_count = M / (16 * MT);
    if (wave >= mg_count * NT) return;         // wave-uniform exit
    int mg = wave / NT;
    int nt = wave % NT;

    int lane = threadIdx.x;                    // wave32
    int mr   = lane & 15;
    int kg   = lane >> 4;                      // K-pair group (0 or 1)
    int col  = nt * 16 + mr;

    const float* blp = WlT + (size_t)col * K + kg * 2;
    const float* brp = WrT + (size_t)col * K + kg * 2;
    const float* ar[MT];
    #pragma unroll
    for (int t = 0; t < MT; ++t)
        ar[t] = A + (size_t)((mg * MT + t) * 16 + mr) * K + kg * 2;

    v8f accL[MT], accR[MT];
    #pragma unroll
    for (int t = 0; t < MT; ++t) { accL[t] = (v8f){}; accR[t] = (v8f){}; }

    for (int k0 = 0; k0 < K; k0 += 4) {
        v2f bl = *(const v2f*)(blp + k0);
        v2f br = *(const v2f*)(brp + k0);
        #pragma unroll
        for (int t = 0; t < MT; ++t) {
            v2f a = *(const v2f*)(ar[t] + k0);
            accL[t] = __builtin_amdgcn_wmma_f32_16x16x4_f32(false, a, false, bl,
                                                            (short)0, accL[t], false, false);
            accR[t] = __builtin_amdgcn_wmma_f32_16x16x4_f32(false, a, false, br,
                                                            (short)0, accR[t], false, false);
        }
    }

    bool colOK = (DOUT == DPAD) || (col < DOUT);
    float bv = colOK ? bias[col] : 0.0f;
    if (colOK) {
        #pragma unroll
        for (int t = 0; t < MT; ++t) {
            #pragma unroll
            for (int v = 0; v < 8; ++v) {
                int row = (mg * MT + t) * 16 + v + kg * 8;
                Z[(size_t)row * DOUT + col] = accL[t][v];
                R[(size_t)row * DOUT + col] = accR[t][v] + bv;
            }
        }
    }
}

// ---------------- Aggregation: out = relu(mean_neighbors(z) + r) ----------------

template<int D>
__global__ void sage_aggregate(const float* __restrict__ z, const float* __restrict__ r,
                               const int* __restrict__ row_ptr, const int* __restrict__ csr_src,
                               const float* __restrict__ deg_inv,
                               float* __restrict__ out, int n) {
    int node = blockIdx.x * (blockDim.x >> 5) + (threadIdx.x >> 5);
    int lane = threadIdx.x & 31;
    if (node >= n) return;
    int beg = row_ptr[node], end = row_ptr[node + 1];
    if constexpr (D == 128) {
        // one float4 per lane -> global_load_b128, 512B coalesced per wave
        float4 acc = make_float4(0.f, 0.f, 0.f, 0.f);
        for (int j = beg; j < end; ++j) {
            int s = csr_src[j];
            float4 v = ((const float4*)(z + (size_t)s * 128))[lane];
            acc.x += v.x; acc.y += v.y; acc.z += v.z; acc.w += v.w;
        }
        float inv = deg_inv[node];
        float4 rv = ((const float4*)(r + (size_t)node * 128))[lane];
        float4 o;
        o.x = fmaxf(acc.x * inv + rv.x, 0.f);
        o.y = fmaxf(acc.y * inv + rv.y, 0.f);
        o.z = fmaxf(acc.z * inv + rv.z, 0.f);
        o.w = fmaxf(acc.w * inv + rv.w, 0.f);
        ((float4*)(out + (size_t)node * 128))[lane] = o;
    } else {
        if (lane < D) {                         // single EXEC set, no per-iter masking
            float acc = 0.f;
            for (int j = beg; j < end; ++j) {
                int s = csr_src[j];
                acc += z[(size_t)s * D + lane];
            }
            float v = acc * deg_inv[node] + r[(size_t)node * D + lane];
            out[(size_t)node * D + lane] = fmaxf(v, 0.f);
        }
    }
}

// ---------------- Host orchestration ----------------

extern "C" void kernel_launch(void* const* d_in, const int* in_sizes, int n_in,
                              void* d_out, int out_size, void* d_ws, size_t ws_size,
                              hipStream_t stream) {
    const float* x  = (const float*)d_in[0];
    const int*   ei = (const int*)d_in[1];
    const float *wl[5], *bl[5], *wr[5];
    for (int i = 0; i < 5; ++i) {
        wl[i] = (const float*)d_in[2 + 3 * i];
        bl[i] = (const float*)d_in[3 + 3 * i];
        wr[i] = (const float*)d_in[4 + 3 * i];
    }

    char*  ws  = (char*)d_ws;
    size_t off = 0;
    auto alloc = [&](size_t bytes) -> void* {
        void* p = ws + off;
        off = (off + bytes + 255) & ~(size_t)255;
        return p;
    };
    int*   cnt     = (int*)alloc((size_t)GN * 4);
    int*   fill    = (int*)alloc((size_t)GN * 4);
    int*   row_ptr = (int*)alloc((size_t)(GN + 1) * 4);
    float* deg_inv = (float*)alloc((size_t)GN * 4);
    int*   csr     = (int*)alloc((size_t)GE * 4);
    float* wlT     = (float*)alloc((size_t)GDH * GDH * 4);
    float* wrT     = (float*)alloc((size_t)GDH * GDH * 4);
    float* z       = (float*)alloc((size_t)GN * GDH * 4);
    float* r       = (float*)alloc((size_t)GN * GDH * 4);
    float* hA      = (float*)alloc((size_t)GN * GDH * 4);
    float* hB      = (float*)alloc((size_t)GN * GDH * 4);

    hipMemsetAsync(cnt,  0, (size_t)GN * 4, stream);
    hipMemsetAsync(fill, 0, (size_t)GN * 4, stream);

    const int* src = ei;
    const int* dst = ei + GE;
    count_kernel<<<(GE + 255) / 256, 256, 0, stream>>>(dst, cnt, GE);
    scan_kernel<<<1, 1024, 0, stream>>>(cnt, row_ptr, deg_inv, GN);
    fill_kernel<<<(GE + 255) / 256, 256, 0, stream>>>(src, dst, row_ptr, fill, csr, GE);

    constexpr int MT = 4;                       // M tiles per wave
    const float* hin  = x;
    float*       hout = hA;
    for (int l = 0; l < 5; ++l) {
        int Dout    = (l == 4) ? 17 : GDH;
        int DoutPad = (l == 4) ? 32 : GDH;
        int tcnt    = DoutPad * GDH;
        transpose_pad<<<(tcnt + 255) / 256, 256, 0, stream>>>(wl[l], wlT, GDH, Dout, DoutPad);
        transpose_pad<<<(tcnt + 255) / 256, 256, 0, stream>>>(wr[l], wrT, GDH, Dout, DoutPad);

        float* dest = (l == 4) ? (float*)d_out : hout;
        dim3 blk(32, 4);
        if (l < 4) {
            int waves  = (GN / (16 * MT)) * (GDH / 16);
            sage_gemm_dual<GDH, GDH, MT><<<(waves + 3) / 4, blk, 0, stream>>>(
                hin, wlT, wrT, bl[l], z, r, GN);
            sage_aggregate<GDH><<<(GN + 7) / 8, 256, 0, stream>>>(
                z, r, row_ptr, csr, deg_inv, dest, GN);
        } else {
            int waves  = (GN / (16 * MT)) * (32 / 16);
            sage_gemm_dual<17, 32, MT><<<(waves + 3) / 4, blk, 0, stream>>>(
                hin, wlT, wrT, bl[l], z, r, GN);
            sage_aggregate<17><<<(GN + 7) / 8, 256, 0, stream>>>(
                z, r, row_ptr, csr, deg_inv, dest, GN);
        }
        hin  = dest;
        hout = (hout == hA) ? hB : hA;
    }
}